// GNN_Layer_18743237279956
// MI455X (gfx1250) — compile-verified
//
#include <hip/hip_runtime.h>
#include <hip/hip_bf16.h>

typedef float v2f __attribute__((ext_vector_type(2)));
typedef float v8f __attribute__((ext_vector_type(8)));

#define EPSV 1e-5f

__device__ __forceinline__ v8f wmma_f32(v2f a, v2f b, v8f c) {
  return __builtin_amdgcn_wmma_f32_16x16x4_f32(false, a, false, b, (short)0, c,
                                               false, false);
}

// ---------------------------------------------------------------------------
// Kernel 1: sq[b,i] = sum_f emb_in[b,i,f]^2   (rows = B*N, F=64)
// ---------------------------------------------------------------------------
__global__ void k_rowsq(const float* __restrict__ emb, float* __restrict__ sq,
                        int rows) {
  int r = blockIdx.x * blockDim.x + threadIdx.x;
  if (r >= rows) return;
  const float4* p = reinterpret_cast<const float4*>(emb + (size_t)r * 64);
  float s = 0.f;
#pragma unroll
  for (int i = 0; i < 16; ++i) {
    float4 v = p[i];
    s += v.x * v.x + v.y * v.y + v.z * v.z + v.w * v.w;
  }
  sq[r] = s;
}

// ---------------------------------------------------------------------------
// Kernel 2: masked node-norm over axis=1: emb_n[b,i,f]
// ---------------------------------------------------------------------------
__global__ void k_norm(const float* __restrict__ emb, const int* __restrict__ nbn,
                       float* __restrict__ emb_n, int N) {
  int b = blockIdx.x >> 6;
  int f = blockIdx.x & 63;
  int nb = nbn[b];
  float cnt = fmaxf((float)nb, 1.0f);
  const float* E = emb + (size_t)b * N * 64;
  float s1 = 0.f, s2 = 0.f;
  for (int i = threadIdx.x; i < N; i += blockDim.x) {
    float x = E[(size_t)i * 64 + f];
    s1 += x;
    s2 += x * x;
  }
  __shared__ float r1[256];
  __shared__ float r2[256];
  r1[threadIdx.x] = s1;
  r2[threadIdx.x] = s2;
  __syncthreads();
  for (int s = 128; s > 0; s >>= 1) {
    if ((int)threadIdx.x < s) {
      r1[threadIdx.x] += r1[threadIdx.x + s];
      r2[threadIdx.x] += r2[threadIdx.x + s];
    }
    __syncthreads();
  }
  float mean = r1[0] / cnt;
  float var = r2[0] / cnt - mean * mean;
  float rstd = rsqrtf(var + EPSV);
  float* EN = emb_n + (size_t)b * N * 64;
  for (int i = threadIdx.x; i < N; i += blockDim.x) {
    float x = E[(size_t)i * 64 + f];
    EN[(size_t)i * 64 + f] = (i < nb) ? (x - mean) * rstd : 0.f;
  }
}

// ---------------------------------------------------------------------------
// Kernel 3: adjacency tile. One wave per 16x64 tile.
// Gram via V_WMMA_F32_16X16X4_F32, epilogue re-mapped through LDS so the
// bandwidth-dominant adj_in/adj_mask/adj streams are all b128.
// ---------------------------------------------------------------------------
__global__ void __launch_bounds__(32) k_adj(
    const float* __restrict__ emb, const float* __restrict__ adj_in,
    const float* __restrict__ adj_mask, const float* __restrict__ sq,
    const float* __restrict__ sigma_p, const float* __restrict__ cw,
    float* __restrict__ adj_out, int N) {
  __shared__ float S[16 * 68];  // 16 rows x 64 cols, pitch 68 (16B-aligned rows)
  int lane = threadIdx.x & 31;
  int lo = lane & 15;
  int hf = lane >> 4;
  int b = blockIdx.z;
  int i0 = blockIdx.x << 4;
  int j0 = blockIdx.y << 6;
  const float* E = emb + (size_t)b * N * 64;

  v8f c[4];
#pragma unroll
  for (int t = 0; t < 4; ++t)
#pragma unroll
    for (int v = 0; v < 8; ++v) c[t][v] = 0.f;

  const float* arow = E + (size_t)(i0 + lo) * 64 + 2 * hf;
#pragma unroll 4
  for (int kb = 0; kb < 64; kb += 4) {
    v2f a = *reinterpret_cast<const v2f*>(arow + kb);
#pragma unroll
    for (int t = 0; t < 4; ++t) {
      v2f bb = *reinterpret_cast<const v2f*>(E + (size_t)(j0 + 16 * t + lo) * 64 +
                                             kb + 2 * hf);
      c[t] = wmma_f32(a, bb, c[t]);
    }
  }

  // C tiles -> LDS (row-major)
#pragma unroll
  for (int t = 0; t < 4; ++t)
#pragma unroll
    for (int v = 0; v < 8; ++v) S[(v + 8 * hf) * 68 + 16 * t + lo] = c[t][v];
  __syncthreads();

  float inv_sigma = 1.0f / sigma_p[0];
  float w0 = cw[0], w1 = cw[1];
  const float* SQ = sq + (size_t)b * N;

#pragma unroll
  for (int it = 0; it < 8; ++it) {
    int idx = it * 32 + lane;  // 0..255
    int row = idx >> 4;        // 0..15
    int q = idx & 15;          // float4 column index
    float4 g = *reinterpret_cast<const float4*>(S + row * 68 + 4 * q);
    float sqi = SQ[i0 + row];
    float4 sj = *reinterpret_cast<const float4*>(SQ + j0 + 4 * q);
    size_t base = ((size_t)b * N + i0 + row) * N + j0 + 4 * q;
    float4 ai = *reinterpret_cast<const float4*>(adj_in + base);
    float4 am = *reinterpret_cast<const float4*>(adj_mask + base);
    float4 r;
    r.x = (w0 * ai.x + w1 * __expf(-fmaxf(sqi + sj.x - 2.f * g.x, 0.f) * inv_sigma)) * am.x;
    r.y = (w0 * ai.y + w1 * __expf(-fmaxf(sqi + sj.y - 2.f * g.y, 0.f) * inv_sigma)) * am.y;
    r.z = (w0 * ai.z + w1 * __expf(-fmaxf(sqi + sj.z - 2.f * g.z, 0.f) * inv_sigma)) * am.z;
    r.w = (w0 * ai.w + w1 * __expf(-fmaxf(sqi + sj.w - 2.f * g.w, 0.f) * inv_sigma)) * am.w;
    *reinterpret_cast<float4*>(adj_out + base) = r;
  }
}

// ---------------------------------------------------------------------------
// Kernel 4: aggregation + MLPs. 128 threads = 4 waves; block owns 64 rows.
// adj 64x64 tile + emb_n 64x64 tile staged in LDS with float4 coalesced
// loads each K-chunk; WMMA inner loop feeds from LDS. Per-wave epilogue:
// deg, [op_deg|op_adj]@conv_W, relu([emb_in|emb_upd]@nu_W), b128 store.
// ---------------------------------------------------------------------------
__global__ void __launch_bounds__(128) k_gnn(
    const float* __restrict__ adj, const float* __restrict__ emb_in,
    const float* __restrict__ emb_n, const float* __restrict__ convW,
    const float* __restrict__ convB, const float* __restrict__ nuW,
    const float* __restrict__ nuB, float* __restrict__ emb_out, int N) {
  __shared__ float smem[8768];            // 35 KB
  float* Btile = smem;                    // [64][68]  emb_n chunk
  float* Atile = smem + 64 * 68;          // [64][68]  adj chunk
  int tid = threadIdx.x;
  int w = tid >> 5;
  int lane = tid & 31;
  int lo = lane & 15;
  int hf = lane >> 4;
  int b = blockIdx.y;
  int ib0 = blockIdx.x << 6;              // block rows
  int i0 = ib0 + (w << 4);                // wave rows
  const float* A = adj + (size_t)b * N * N;
  const float* EN = emb_n + (size_t)b * N * 64;
  const float* EI = emb_in + (size_t)b * N * 64;

  v8f c[4];
#pragma unroll
  for (int t = 0; t < 4; ++t)
#pragma unroll
    for (int v = 0; v < 8; ++v) c[t][v] = 0.f;
  float dacc = 0.f;

  for (int kb = 0; kb < N; kb += 64) {
    __syncthreads();
#pragma unroll
    for (int it = 0; it < 8; ++it) {
      int idx = it * 128 + tid;  // 0..1023
      int r = idx >> 4;          // 0..63
      int q = idx & 15;          // float4 col
      *reinterpret_cast<float4*>(Btile + r * 68 + 4 * q) =
          *reinterpret_cast<const float4*>(EN + (size_t)(kb + r) * 64 + 4 * q);
      *reinterpret_cast<float4*>(Atile + r * 68 + 4 * q) =
          *reinterpret_cast<const float4*>(A + (size_t)(ib0 + r) * N + kb + 4 * q);
    }
    __syncthreads();
    const float* ar = Atile + (w * 16 + lo) * 68 + 2 * hf;
#pragma unroll
    for (int kk = 0; kk < 64; kk += 4) {
      v2f a = *reinterpret_cast<const v2f*>(ar + kk);
      dacc += a.x + a.y;
      const float* br = Btile + (kk + 2 * hf) * 68 + lo;
#pragma unroll
      for (int t = 0; t < 4; ++t) {
        v2f bb;
        bb.x = br[16 * t];
        bb.y = br[68 + 16 * t];
        c[t] = wmma_f32(a, bb, c[t]);
      }
    }
  }
  __syncthreads();

  float* H = smem + w * 2112;             // per-wave [16][132]
  float* degs = smem + 8448 + w * 16;
  float deg = dacc + __shfl_xor(dacc, 16, 32);
  if (hf == 0) degs[lo] = deg;
  __syncthreads();

  // H = [op_deg | op_adj]
#pragma unroll
  for (int it = 0; it < 8; ++it) {
    int idx = it * 32 + lane;
    int m = idx >> 4, q = idx & 15;
    float4 e = *reinterpret_cast<const float4*>(EN + (size_t)(i0 + m) * 64 + 4 * q);
    float dm = degs[m];
    float4 o;
    o.x = dm * e.x; o.y = dm * e.y; o.z = dm * e.z; o.w = dm * e.w;
    *reinterpret_cast<float4*>(H + m * 132 + 4 * q) = o;
  }
#pragma unroll
  for (int t = 0; t < 4; ++t)
#pragma unroll
    for (int v = 0; v < 8; ++v)
      H[(v + 8 * hf) * 132 + 64 + 16 * t + lo] = c[t][v];
  __syncthreads();

  // emb_upd = H @ conv_W + conv_b
  v8f u[4];
#pragma unroll
  for (int t = 0; t < 4; ++t)
#pragma unroll
    for (int v = 0; v < 8; ++v) u[t][v] = 0.f;
#pragma unroll 4
  for (int kb = 0; kb < 128; kb += 4) {
    v2f a = *reinterpret_cast<const v2f*>(H + lo * 132 + kb + 2 * hf);
#pragma unroll
    for (int t = 0; t < 4; ++t) {
      v2f bb;
      bb.x = convW[(size_t)(kb + 2 * hf) * 64 + 16 * t + lo];
      bb.y = convW[(size_t)(kb + 2 * hf + 1) * 64 + 16 * t + lo];
      u[t] = wmma_f32(a, bb, u[t]);
    }
  }
  __syncthreads();

  // H = [emb_in | emb_upd]
#pragma unroll
  for (int it = 0; it < 8; ++it) {
    int idx = it * 32 + lane;
    int m = idx >> 4, q = idx & 15;
    *reinterpret_cast<float4*>(H + m * 132 + 4 * q) =
        *reinterpret_cast<const float4*>(EI + (size_t)(i0 + m) * 64 + 4 * q);
  }
#pragma unroll
  for (int t = 0; t < 4; ++t) {
    float bias = convB[16 * t + lo];
#pragma unroll
    for (int v = 0; v < 8; ++v)
      H[(v + 8 * hf) * 132 + 64 + 16 * t + lo] = u[t][v] + bias;
  }
  __syncthreads();

  // emb = relu(H @ nu_W + nu_b)
  v8f z[4];
#pragma unroll
  for (int t = 0; t < 4; ++t)
#pragma unroll
    for (int v = 0; v < 8; ++v) z[t][v] = 0.f;
#pragma unroll 4
  for (int kb = 0; kb < 128; kb += 4) {
    v2f a = *reinterpret_cast<const v2f*>(H + lo * 132 + kb + 2 * hf);
#pragma unroll
    for (int t = 0; t < 4; ++t) {
      v2f bb;
      bb.x = nuW[(size_t)(kb + 2 * hf) * 64 + 16 * t + lo];
      bb.y = nuW[(size_t)(kb + 2 * hf + 1) * 64 + 16 * t + lo];
      z[t] = wmma_f32(a, bb, z[t]);
    }
  }
  __syncthreads();
#pragma unroll
  for (int t = 0; t < 4; ++t) {
    float bias = nuB[16 * t + lo];
#pragma unroll
    for (int v = 0; v < 8; ++v)
      H[(v + 8 * hf) * 132 + 16 * t + lo] = fmaxf(z[t][v] + bias, 0.f);
  }
  __syncthreads();
#pragma unroll
  for (int it = 0; it < 8; ++it) {
    int idx = it * 32 + lane;
    int m = idx >> 4, q = idx & 15;
    *reinterpret_cast<float4*>(emb_out + ((size_t)b * N + i0 + m) * 64 + 4 * q) =
        *reinterpret_cast<const float4*>(H + m * 132 + 4 * q);
  }
}

// ---------------------------------------------------------------------------
extern "C" void kernel_launch(void* const* d_in, const int* in_sizes, int n_in,
                              void* d_out, int out_size, void* d_ws,
                              size_t ws_size, hipStream_t stream) {
  const float* emb_in = (const float*)d_in[0];
  const float* adj_in = (const float*)d_in[1];
  const float* adj_mask = (const float*)d_in[2];
  const int* nbn = (const int*)d_in[3];
  const float* sigma = (const float*)d_in[4];
  const float* cw = (const float*)d_in[5];
  const float* convW = (const float*)d_in[6];
  const float* convB = (const float*)d_in[7];
  const float* nuW = (const float*)d_in[8];
  const float* nuB = (const float*)d_in[9];

  const int F = 64;
  int B = in_sizes[3];
  int N = in_sizes[0] / (B * F);

  float* emb_out = (float*)d_out;                     // (B,N,F)
  float* adj_out = (float*)d_out + (size_t)B * N * F; // (B,N,N)

  float* emb_n = (float*)d_ws;                        // B*N*F floats
  float* sq = emb_n + (size_t)B * N * F;              // B*N floats

  int rows = B * N;
  k_rowsq<<<(rows + 255) / 256, 256, 0, stream>>>(emb_in, sq, rows);
  k_norm<<<B * F, 256, 0, stream>>>(emb_in, nbn, emb_n, N);

  dim3 g2(N / 16, N / 64, B);
  k_adj<<<g2, 32, 0, stream>>>(emb_in, adj_in, adj_mask, sq, sigma, cw, adj_out,
                               N);

  dim3 g3(N / 64, B, 1);
  k_gnn<<<g3, 128, 0, stream>>>(adj_out, emb_in, emb_n, convW, convB, nuW, nuB,
                                emb_out, N);
}